// Adder2D_21131239097101
// MI455X (gfx1250) — compile-verified
//
#include <hip/hip_runtime.h>
#include <hip/hip_bf16.h>

// AdderNet L1-conv: out[n,co,h,w] = -sum_{ci,kh,kw} |x[n,ci,h+kh-1,w+kw-1] - w[co,ci,kh,kw]|
// N=16, Ci=Co=64, H=W=32, K=3, pad=1, stride=1.

#define KK 3
#define CI 64
#define CO 64
#define HH 32
#define WW 32
#define NB 16
#define TB 16          // 16x16 spatial tile per block
#define CO_TILE 16     // co's per block
#define CI_CHUNK 4     // ci's staged per TDM (double buffered)
#define NCHUNK (CI / CI_CHUNK)
#define PATCH_ROWS 17  // valid rows in halo'd tile (32-image, 16-tile, halo 1 -> always 17)
#define XS_ELEMS (CI_CHUNK * PATCH_ROWS * WW)

typedef unsigned int uint32x4 __attribute__((ext_vector_type(4)));
typedef int          int32x8  __attribute__((ext_vector_type(8)));
typedef int          int32x4  __attribute__((ext_vector_type(4)));

#ifndef __has_builtin
#define __has_builtin(x) 0
#endif
#if defined(__HIP_DEVICE_COMPILE__) && __has_builtin(__builtin_amdgcn_tensor_load_to_lds) && \
    __has_builtin(__builtin_amdgcn_s_wait_tensorcnt)
#define USE_TDM 1
#else
#define USE_TDM 0
#endif

#if USE_TDM
// Issue one TDM: 3D tile (32 cols x 17 rows x CI_CHUNK ci) -> contiguous LDS at lds_off.
__device__ __forceinline__ void tdm_issue_chunk(const float* gsrc, unsigned lds_off) {
    unsigned long long ga = (unsigned long long)(size_t)gsrc;
    uint32x4 g0;
    g0.x = 1u;                                               // count=1 valid descriptor
    g0.y = lds_off;                                          // lds_addr (bytes)
    g0.z = (unsigned)ga;                                     // global_addr[31:0]
    g0.w = (unsigned)((ga >> 32) & 0x1FFFFFFu) | (2u << 30); // addr[56:32] | type=2
    int32x8 g1;
    g1[0] = (int)(2u << 16);                                 // data_size=4B; no iterate/pad/mask
    g1[1] = (int)(((unsigned)WW & 0xFFFFu) << 16);           // tensor_dim0 = 32 (lo16 @63:48)
    g1[2] = (int)(((unsigned)HH & 0xFFFFu) << 16);           // tensor_dim1 = 32 (lo16 @95:80)
    g1[3] = (int)(((unsigned)WW & 0xFFFFu) << 16);           // tile_dim0 = 32 (@127:112)
    g1[4] = (int)(((unsigned)CI_CHUNK << 16) | PATCH_ROWS);  // tile_dim2=4 | tile_dim1=17
    g1[5] = WW;                                              // tensor_dim0_stride = 32
    g1[6] = (int)(((unsigned)(HH * WW) & 0xFFFFu) << 16);    // tensor_dim1_stride = 1024 (lo16)
    g1[7] = 0;
    int32x4 g2; g2[0] = CI_CHUNK; g2[1] = 0; g2[2] = 0; g2[3] = 0;  // tensor_dim2 = 4
    int32x4 g3; g3[0] = 0; g3[1] = 0; g3[2] = 0; g3[3] = 0;
    int32x8 g4; g4[0] = 0; g4[1] = 0; g4[2] = 0; g4[3] = 0;
    g4[4] = 0; g4[5] = 0; g4[6] = 0; g4[7] = 0;
    __builtin_amdgcn_tensor_load_to_lds(g0, g1, g2, g3, g4, 0);
}
#endif

__global__ __launch_bounds__(256) void adder2d_kernel(const float* __restrict__ x,
                                                      const float* __restrict__ w,
                                                      float* __restrict__ out) {
    __shared__ float wl[CO_TILE * CI * 9];   // 36 KB: weights for this co tile
    __shared__ float xs[2][XS_ELEMS];        // 2 x 8.5 KB: double-buffered x patch

    const int t    = threadIdx.x;
    const int tile = blockIdx.x;            // 0..3 spatial tile
    const int co_t = blockIdx.y;            // 0..3 co tile
    const int n    = blockIdx.z;            // 0..15 batch
    const int tr = tile >> 1, tc = tile & 1;
    const int R0 = tr * TB, C0 = tc * TB;
    const int r_start = (R0 == 0) ? 0 : (R0 - 1);   // first valid global row in patch

    // ---- stage weights for this co tile (contiguous slice of w) ----
    {
        const float* wsrc = w + (size_t)co_t * (CO_TILE * CI * 9);
        for (int i = t; i < CO_TILE * CI * 9; i += 256) wl[i] = wsrc[i];
    }

    // ---- thread tiling: 4 co x 4 adjacent cols per thread ----
    const int pg = t & 63;                  // 64 pixel groups
    const int g  = t >> 6;                  // co subgroup 0..3 (uniform per wave -> LDS broadcast)
    const int r  = pg >> 2;                 // row in tile 0..15
    const int cb = (pg & 3) * 4;            // col base in tile {0,4,8,12}
    const int rowbase = R0 + r - 1 - r_start;   // patch row of kh=0 tap
    const int cwin    = C0 + cb - 1;            // global col of window start

    float acc[4][4];
    #pragma unroll
    for (int j = 0; j < 4; ++j)
        #pragma unroll
        for (int p = 0; p < 4; ++p) acc[j][p] = 0.0f;

    const float* xbase = x + (size_t)n * CI * (HH * WW) + r_start * WW;  // chunk 0 source
#if USE_TDM
    const unsigned lds_xs0 = (unsigned)(size_t)(void*)&xs[0][0];  // LDS byte offset of buffer 0
    // ---- prologue: kick off chunk 0 into buffer 0 ----
    if (t < 32) tdm_issue_chunk(xbase, lds_xs0);
#else
    for (int i = t; i < XS_ELEMS; i += 256) {
        int ci  = i / (PATCH_ROWS * WW);
        int rem = i - ci * (PATCH_ROWS * WW);
        xs[0][i] = xbase[(size_t)ci * (HH * WW) + rem];
    }
#endif
    __syncthreads();    // weights staged (and fallback chunk 0 visible)

    for (int k = 0; k < NCHUNK; ++k) {
        const int b = k & 1;
#if USE_TDM
        if (t < 32) {   // wave 0: prefetch next chunk, then wait for current (in-order TDM)
            if (k + 1 < NCHUNK) {
                tdm_issue_chunk(xbase + (size_t)(k + 1) * CI_CHUNK * (HH * WW),
                                lds_xs0 + (unsigned)(((k + 1) & 1) * XS_ELEMS * 4));
                __builtin_amdgcn_s_wait_tensorcnt(1);   // chunk k complete, k+1 in flight
            } else {
                __builtin_amdgcn_s_wait_tensorcnt(0);   // last chunk complete
            }
        }
        __syncthreads();    // buffer b now valid for all waves
#endif

        for (int ci = 0; ci < CI_CHUNK; ++ci) {
            const int ciG = k * CI_CHUNK + ci;

            // weights: 36 wave-uniform LDS reads (broadcast, b128-mergeable)
            float wreg[4][9];
            #pragma unroll
            for (int j = 0; j < 4; ++j)
                #pragma unroll
                for (int q = 0; q < 9; ++q)
                    wreg[j][q] = wl[((g * 4 + j) * CI + ciG) * 9 + q];

            // x window 3x6 with zero-padding masks (pad contributes x=0 -> |w|)
            float xw[3][6];
            #pragma unroll
            for (int kh = 0; kh < 3; ++kh) {
                int  pr  = rowbase + kh;
                bool rok = (unsigned)pr < (unsigned)PATCH_ROWS;
                int  prc = rok ? pr : 0;
                #pragma unroll
                for (int m = 0; m < 6; ++m) {
                    int  gc  = cwin + m;
                    bool cok = (unsigned)gc < (unsigned)WW;
                    int  gcc = cok ? gc : 0;
                    float v  = xs[b][(ci * PATCH_ROWS + prc) * WW + gcc];
                    xw[kh][m] = (rok && cok) ? v : 0.0f;
                }
            }

            // 4co x 4px x 9 taps: 2 VALU/tap (v_sub + v_sub with |.| modifier)
            #pragma unroll
            for (int j = 0; j < 4; ++j)
                #pragma unroll
                for (int kh = 0; kh < 3; ++kh)
                    #pragma unroll
                    for (int kw = 0; kw < 3; ++kw) {
                        float wv = wreg[j][kh * 3 + kw];
                        #pragma unroll
                        for (int p = 0; p < 4; ++p)
                            acc[j][p] -= fabsf(xw[kh][p + kw] - wv);
                    }
        }
        __syncthreads();    // reads of buffer b done; next iter's TDM may overwrite it

#if !USE_TDM
        if (k + 1 < NCHUNK) {   // fallback: stage next chunk synchronously
            const float* src = xbase + (size_t)(k + 1) * CI_CHUNK * (HH * WW);
            float* dst = xs[(k + 1) & 1];
            for (int i = t; i < XS_ELEMS; i += 256) {
                int ci  = i / (PATCH_ROWS * WW);
                int rem = i - ci * (PATCH_ROWS * WW);
                dst[i] = src[(size_t)ci * (HH * WW) + rem];
            }
            __syncthreads();
        }
#endif
    }

    // ---- store: 4 contiguous floats per co -> b128 stores ----
    #pragma unroll
    for (int j = 0; j < 4; ++j) {
        int co = co_t * CO_TILE + g * 4 + j;
        size_t o = (((size_t)n * CO + co) * HH + (R0 + r)) * WW + (C0 + cb);
        float4 v = make_float4(acc[j][0], acc[j][1], acc[j][2], acc[j][3]);
        *reinterpret_cast<float4*>(out + o) = v;
    }
}

extern "C" void kernel_launch(void* const* d_in, const int* in_sizes, int n_in,
                              void* d_out, int out_size, void* d_ws, size_t ws_size,
                              hipStream_t stream) {
    const float* x = (const float*)d_in[0];   // [16,64,32,32]
    const float* w = (const float*)d_in[1];   // [64,64,3,3]
    float* out = (float*)d_out;               // [16,64,32,32]
    (void)in_sizes; (void)n_in; (void)out_size; (void)d_ws; (void)ws_size;

    dim3 grid(4, 4, NB);   // spatial tiles, co tiles, batch
    dim3 block(256);
    adder2d_kernel<<<grid, block, 0, stream>>>(x, w, out);
}